// RGCNLayer_87943750353108
// MI455X (gfx1250) — compile-verified
//
#include <hip/hip_runtime.h>
#include <hip/hip_bf16.h>
#include <math.h>

typedef float v2f __attribute__((ext_vector_type(2)));
typedef float v8f __attribute__((ext_vector_type(8)));

#define IN_F     64
#define OUT_F    64
#define SKIP_DIM 64
#define N_RELS   10
#define N_BASES  5
#define TILE_M   128   // node tile for projection GEMM
#define TILE4_M  32    // node tile for skip GEMM

// ---------------------------------------------------------------------------
// Kernel 1: basis decomposition. Faithful to the torch view/matmul/view chain:
// the two reshapes are flat reinterpretations, so
//   w_flat[i*640 + r*64 + o] = sum_b w_comp[r,b] * weight_flat[i*320 + b*64 + o]
// and the final [R,in,out] view indexes the SAME flat buffer as r*4096+i*64+o.
// Output goes directly into the tail of d_out (it is a returned tensor) and is
// re-read by the projection GEMM.
// ---------------------------------------------------------------------------
__global__ __launch_bounds__(256) void k_basis(const float* __restrict__ weight,
                                               const float* __restrict__ w_comp,
                                               float* __restrict__ w_out) {
  int f = blockIdx.x * 256 + threadIdx.x;
  if (f >= N_RELS * IN_F * OUT_F) return;
  int i   = f / (N_RELS * OUT_F);
  int rem = f % (N_RELS * OUT_F);
  int r   = rem / OUT_F;
  int o   = rem % OUT_F;
  float acc = 0.f;
#pragma unroll
  for (int b = 0; b < N_BASES; ++b)
    acc += w_comp[r * N_BASES + b] * weight[i * (N_BASES * OUT_F) + b * OUT_F + o];
  w_out[f] = acc;
}

// ---------------------------------------------------------------------------
// Kernel 2: h_proj[n,r,:] = h[n,:] @ w_r  (WMMA f32 16x16x4), fused gate logits
// g_all[n,r] = h[n,:] . gate_w[r,:].
// Block = 256 thr (8 waves) handles a 128-node tile for one relation.
// LDS: h tile 128x(64+4), w_r 64x(64+4)  => ~51 KB.
// ---------------------------------------------------------------------------
__global__ __launch_bounds__(256) void k_proj(const float* __restrict__ h,
                                              const float* __restrict__ w,      // [R*64*64] flat
                                              const float* __restrict__ gate_w, // [R*64]
                                              float* __restrict__ h_proj,       // [N,R,64]
                                              float* __restrict__ g_all,        // [N,R]
                                              int n_nodes) {
  __shared__ float hs[TILE_M][IN_F + 4];
  __shared__ float ws[IN_F][OUT_F + 4];
  const int tid   = threadIdx.x;
  const int r     = blockIdx.y;
  const int node0 = blockIdx.x * TILE_M;

  // stage h tile (float4, zero-padded past n_nodes): 2048 float4 / 256 thr = 8 each
#pragma unroll
  for (int t = 0; t < 8; ++t) {
    int idx = t * 256 + tid;
    int row = idx >> 4;
    int c4  = (idx & 15) << 2;
    float4 v = make_float4(0.f, 0.f, 0.f, 0.f);
    int n = node0 + row;
    if (n < n_nodes) v = *(const float4*)(h + (size_t)n * IN_F + c4);
    hs[row][c4 + 0] = v.x; hs[row][c4 + 1] = v.y;
    hs[row][c4 + 2] = v.z; hs[row][c4 + 3] = v.w;
  }
  // stage w_r: 1024 float4 / 256 thr = 4 each
#pragma unroll
  for (int t = 0; t < 4; ++t) {
    int idx = t * 256 + tid;
    int row = idx >> 4;
    int c4  = (idx & 15) << 2;
    float4 v = *(const float4*)(w + (size_t)r * IN_F * OUT_F + row * OUT_F + c4);
    ws[row][c4 + 0] = v.x; ws[row][c4 + 1] = v.y;
    ws[row][c4 + 2] = v.z; ws[row][c4 + 3] = v.w;
  }
  __syncthreads();

  // fused gate logits (threads 0..127, one node each; reads LDS, writes global)
  if (tid < TILE_M) {
    int n = node0 + tid;
    if (n < n_nodes) {
      float acc = 0.f;
#pragma unroll
      for (int i = 0; i < IN_F; ++i) acc += hs[tid][i] * gate_w[r * IN_F + i];
      g_all[(size_t)n * N_RELS + r] = acc;
    }
  }

  // WMMA: wave 'mt' (0..7) owns rows [mt*16, mt*16+16); loops 4 col tiles.
  const int wave = tid >> 5, lane = tid & 31;
  const int half = lane >> 4, l16 = lane & 15;
  const int mt = wave;
  for (int nt = 0; nt < 4; ++nt) {
    v8f acc = {0.f, 0.f, 0.f, 0.f, 0.f, 0.f, 0.f, 0.f};
#pragma unroll
    for (int k = 0; k < 16; ++k) {
      int kk = k * 4 + half * 2;                 // A 16x4: lanes<16 K={0,1}, lanes>=16 K={2,3}
      v2f a; a.x = hs[mt * 16 + l16][kk]; a.y = hs[mt * 16 + l16][kk + 1];
      v2f b; b.x = ws[kk][nt * 16 + l16]; b.y = ws[kk + 1][nt * 16 + l16];
      acc = __builtin_amdgcn_wmma_f32_16x16x4_f32(false, a, false, b,
                                                  (short)0, acc, false, false);
    }
#pragma unroll
    for (int v = 0; v < 8; ++v) {
      int m = mt * 16 + v + half * 8;            // D: VGPR v -> rows v / v+8
      int n = node0 + m;
      if (n < n_nodes)
        h_proj[((size_t)n * N_RELS + r) * OUT_F + nt * 16 + l16] = acc[v];
    }
  }
}

// ---------------------------------------------------------------------------
// Kernel 3: edge gather/scale/scatter. One wave per edge; each lane owns 2 of
// the 64 output floats. h_proj (128 MB) + agg (12.8 MB) are L2-resident on
// MI455X (192 MB L2), so the gathers and fp32 atomics stay on-chip.
// ---------------------------------------------------------------------------
__global__ __launch_bounds__(256) void k_edge(const float* __restrict__ h_proj,
                                              const float* __restrict__ g_all,
                                              const float* __restrict__ norm,
                                              const int* __restrict__ esrc,
                                              const int* __restrict__ edst,
                                              const int* __restrict__ erel,
                                              const float* __restrict__ bias_gate,
                                              float* __restrict__ agg,
                                              int n_edges) {
  int gid = blockIdx.x * 256 + threadIdx.x;
  int e = gid >> 5;
  if (e >= n_edges) return;
  int lane = gid & 31;
  int src = esrc[e], dst = edst[e], rt = erel[e];
  float g    = g_all[(size_t)src * N_RELS + rt] + bias_gate[0];
  float gate = 1.f / (1.f + __expf(-g));
  float s    = norm[e] * gate;
  const float2 p = *(const float2*)(h_proj + ((size_t)src * N_RELS + rt) * OUT_F + lane * 2);
  float* dp = agg + (size_t)dst * OUT_F + lane * 2;
  unsafeAtomicAdd(dp + 0, p.x * s);   // -> global_atomic_add_f32
  unsafeAtomicAdd(dp + 1, p.y * s);
}

// ---------------------------------------------------------------------------
// Kernel 4: hh = relu(agg+bias); out = relu([hh | h_skip] @ skip_W^T + skip_b)
// WMMA f32 16x16x4, K=128. Block = 256 thr handles 32 nodes; Bt (=skip_W^T)
// and the concatenated A tile are built in LDS (~50 KB).
// ---------------------------------------------------------------------------
__global__ __launch_bounds__(256) void k_skip(const float* __restrict__ agg,
                                              const float* __restrict__ bias,
                                              const float* __restrict__ h_skip,
                                              const float* __restrict__ skip_W, // [64,128]
                                              const float* __restrict__ skip_b,
                                              float* __restrict__ out,
                                              int n_nodes) {
  __shared__ float As[TILE4_M][128 + 4];
  __shared__ float Bt[128][OUT_F + 2];
  const int tid   = threadIdx.x;
  const int node0 = blockIdx.x * TILE4_M;

  // Bt[k][o] = skip_W[o*128+k] : 8192 elems, coalesced on k
#pragma unroll
  for (int t = 0; t < 32; ++t) {
    int idx = t * 256 + tid;
    int o = idx >> 7;
    int k = idx & 127;
    Bt[k][o] = skip_W[(size_t)o * 128 + k];
  }
  // A tile: cols 0..63 = relu(agg+bias), 64..127 = h_skip
#pragma unroll
  for (int t = 0; t < 16; ++t) {
    int idx = t * 256 + tid;
    int row = idx >> 7;
    int c   = idx & 127;
    int n   = node0 + row;
    float v = 0.f;
    if (n < n_nodes) {
      if (c < OUT_F) v = fmaxf(agg[(size_t)n * OUT_F + c] + bias[c], 0.f);
      else           v = h_skip[(size_t)n * SKIP_DIM + (c - OUT_F)];
    }
    As[row][c] = v;
  }
  __syncthreads();

  const int wave = tid >> 5, lane = tid & 31;
  const int half = lane >> 4, l16 = lane & 15;
  const int mt = wave >> 2;   // 0..1 (2 row tiles)
  const int nt = wave & 3;    // 0..3 (4 col tiles)
  v8f acc = {0.f, 0.f, 0.f, 0.f, 0.f, 0.f, 0.f, 0.f};
#pragma unroll
  for (int k = 0; k < 32; ++k) {
    int kk = k * 4 + half * 2;
    v2f a; a.x = As[mt * 16 + l16][kk]; a.y = As[mt * 16 + l16][kk + 1];
    v2f b; b.x = Bt[kk][nt * 16 + l16]; b.y = Bt[kk + 1][nt * 16 + l16];
    acc = __builtin_amdgcn_wmma_f32_16x16x4_f32(false, a, false, b,
                                                (short)0, acc, false, false);
  }
#pragma unroll
  for (int v = 0; v < 8; ++v) {
    int m   = mt * 16 + v + half * 8;
    int n   = node0 + m;
    int col = nt * 16 + l16;
    if (n < n_nodes)
      out[(size_t)n * OUT_F + col] = fmaxf(acc[v] + skip_b[col], 0.f);
  }
}

// ---------------------------------------------------------------------------
extern "C" void kernel_launch(void* const* d_in, const int* in_sizes, int n_in,
                              void* d_out, int out_size, void* d_ws, size_t ws_size,
                              hipStream_t stream) {
  const float* h         = (const float*)d_in[0];
  const float* h_skip    = (const float*)d_in[1];
  const float* norm      = (const float*)d_in[2];
  const int*   esrc      = (const int*)d_in[3];
  const int*   edst      = (const int*)d_in[4];
  const int*   erel      = (const int*)d_in[5];
  /* d_in[6] = layer_num (==2, skip branch always taken in reference) */
  const float* weight    = (const float*)d_in[7];
  const float* w_comp    = (const float*)d_in[8];
  const float* bias      = (const float*)d_in[9];
  const float* bias_gate = (const float*)d_in[10];
  const float* gate_w    = (const float*)d_in[11];
  const float* skip_W    = (const float*)d_in[12];
  const float* skip_b    = (const float*)d_in[13];

  const int n_nodes = in_sizes[0] / IN_F;
  const int n_edges = in_sizes[3];

  float* out   = (float*)d_out;
  float* w_out = out + (size_t)n_nodes * OUT_F;      // second returned tensor [R,in,out]

  float* ws     = (float*)d_ws;
  float* h_proj = ws;                                            // N*R*64 f32 (128 MB)
  float* g_all  = h_proj + (size_t)n_nodes * N_RELS * OUT_F;     // N*R     f32 (2 MB)
  float* agg    = g_all  + (size_t)n_nodes * N_RELS;             // N*64    f32 (12.8 MB)

  hipMemsetAsync(agg, 0, (size_t)n_nodes * OUT_F * sizeof(float), stream);

  k_basis<<<(N_RELS * IN_F * OUT_F + 255) / 256, 256, 0, stream>>>(weight, w_comp, w_out);

  dim3 g2((n_nodes + TILE_M - 1) / TILE_M, N_RELS);
  k_proj<<<g2, 256, 0, stream>>>(h, w_out, gate_w, h_proj, g_all, n_nodes);

  int eblocks = (n_edges + 7) / 8;   // 8 waves/block, 1 edge per wave
  k_edge<<<eblocks, 256, 0, stream>>>(h_proj, g_all, norm, esrc, edst, erel,
                                      bias_gate, agg, n_edges);

  k_skip<<<(n_nodes + TILE4_M - 1) / TILE4_M, 256, 0, stream>>>(
      agg, bias, h_skip, skip_W, skip_b, out, n_nodes);
}